// MultiTaskGCN_link_25340307046431
// MI455X (gfx1250) — compile-verified
//
#include <hip/hip_runtime.h>
#include <hip/hip_bf16.h>
#include <stdint.h>

typedef __attribute__((ext_vector_type(2))) float v2f;
typedef __attribute__((ext_vector_type(8))) float v8f;
typedef __attribute__((ext_vector_type(2))) int   v2i_t;

// address-space-qualified views of the b64 async payload type
typedef __attribute__((address_space(1))) v2i_t as1_v2i;   // global
typedef __attribute__((address_space(3))) v2i_t as3_v2i;   // LDS

#define FEAT_IN  512
#define FEAT_H   16
#define FEAT_OUT 64
#define GK       16      // K-chunk staged per async round
#define XS_STRIDE 18     // padded row stride (floats): conflict-free ds_load_b64

#if __has_builtin(__builtin_amdgcn_global_load_async_to_lds_b64)
#define HAVE_ASYNC 1
#else
#define HAVE_ASYNC 0
#endif

__device__ __forceinline__ void async_b64(const float* gp, float* lp) {
#if HAVE_ASYNC
    __builtin_amdgcn_global_load_async_to_lds_b64(
        (as1_v2i*)(v2i_t*)(gp),
        (as3_v2i*)(v2i_t*)(lp), 0, 0);
#else
    *(v2f*)lp = *(const v2f*)gp;   // synchronous fallback
#endif
}

template <int n>
__device__ __forceinline__ void wait_async() {
#if HAVE_ASYNC
#if __has_builtin(__builtin_amdgcn_s_wait_asynccnt)
    __builtin_amdgcn_s_wait_asynccnt(n);
#else
    asm volatile("s_wait_asynccnt %0" :: "i"(n) : "memory");
#endif
#endif
}

// ---------------------------------------------------------------------------
__global__ void init_deg_kernel(float* __restrict__ deg, int N) {
    int i = blockIdx.x * blockDim.x + threadIdx.x;
    if (i < N) deg[i] = 1.0f;                       // self loop
}

__global__ void deg_edge_kernel(const int* __restrict__ ei, float* __restrict__ deg, int E) {
    int e = blockIdx.x * blockDim.x + threadIdx.x;
    if (e < E) unsafeAtomicAdd(&deg[ei[E + e]], 1.0f);
}

__global__ void dinv_kernel(const float* __restrict__ deg, float* __restrict__ dinv, int N) {
    int i = blockIdx.x * blockDim.x + threadIdx.x;
    if (i < N) dinv[i] = rsqrtf(deg[i]);
}

// dst[i] = bias[i % len]  (bias folded into the scatter accumulator init)
__global__ void init_rep_kernel(float* __restrict__ dst, const float* __restrict__ bias,
                                int len, long long total) {
    long long i = (long long)blockIdx.x * blockDim.x + threadIdx.x;
    if (i < total) dst[i] = bias[i % len];
}

// ---------------------------------------------------------------------------
// h0 = x @ W1  (100000x512 @ 512x16), f32 WMMA 16x16x4.
// One 16-row tile per wave; x streamed through LDS with async-to-LDS double
// buffering (4 b64 async ops per 16-K chunk, s_wait_asynccnt 4 overlap).
__global__ void gemm1_kernel(const float* __restrict__ x, const float* __restrict__ W1,
                             float* __restrict__ h0, int N) {
    __shared__ float w1s[FEAT_IN * FEAT_H];                 // 32 KB
    __shared__ float xs[8][2][16 * XS_STRIDE];              // 18 KB (per-wave bufs)
    for (int i = threadIdx.x; i < FEAT_IN * FEAT_H; i += blockDim.x) w1s[i] = W1[i];
    __syncthreads();

    const int lane = threadIdx.x & 31;
    const int wid  = threadIdx.x >> 5;
    const int tile = blockIdx.x * (blockDim.x >> 5) + wid;
    const int m    = lane & 15;          // A row / C column
    const int half = lane >> 4;
    // staging mapping: 8 lanes x 2 floats cover one row's 16-float chunk
    const int srow = lane >> 3;          // 0..3 (stepped by 4)
    const int seg  = (lane & 7) * 2;     // float offset within chunk

    // stage chunk c into buffer buf (rows clamp to N-1; invalid C rows unstored)
    auto stage = [&](int c, int buf) {
        const int k0 = c * GK;
#pragma unroll
        for (int i = 0; i < 4; ++i) {
            int r = srow + 4 * i;
            int grow = tile * 16 + r; if (grow >= N) grow = N - 1;
            async_b64(x + (long long)grow * FEAT_IN + k0 + seg,
                      &xs[wid][buf][r * XS_STRIDE + seg]);
        }
    };

    v8f acc = {};
    stage(0, 0);
    const int NCHUNK = FEAT_IN / GK;                        // 32
    for (int c = 0; c < NCHUNK; ++c) {
        const int buf = c & 1;
        if (c + 1 < NCHUNK) { stage(c + 1, buf ^ 1); wait_async<4>(); }
        else                {                        wait_async<0>(); }
        const int k0 = c * GK;
#pragma unroll
        for (int kk = 0; kk < GK; kk += 4) {
            v2f a = *(const v2f*)&xs[wid][buf][m * XS_STRIDE + kk + 2 * half];
            const int kb = (k0 + kk + 2 * half) * FEAT_H + m;
            v2f b = { w1s[kb], w1s[kb + FEAT_H] };
            acc = __builtin_amdgcn_wmma_f32_16x16x4_f32(false, a, false, b,
                                                        (short)0, acc, false, false);
        }
    }
#pragma unroll
    for (int r = 0; r < 8; ++r) {          // C: vgpr r -> M=r (lanes 0-15), M=8+r (16-31)
        int orow = tile * 16 + r + 8 * half;
        if (orow < N) h0[(long long)orow * FEAT_H + m] = acc[r];
    }
}

// z0 = relu(hagg) @ W2  (100000x16 @ 16x64); ReLU fused into A load.
__global__ void gemm2_kernel(const float* __restrict__ hagg, const float* __restrict__ W2,
                             float* __restrict__ z0, int N) {
    __shared__ float w2s[FEAT_H * FEAT_OUT];                // 4 KB
    for (int i = threadIdx.x; i < FEAT_H * FEAT_OUT; i += blockDim.x) w2s[i] = W2[i];
    __syncthreads();

    const int lane = threadIdx.x & 31;
    const int tile = blockIdx.x * (blockDim.x >> 5) + (threadIdx.x >> 5);
    const int m    = lane & 15;
    const int half = lane >> 4;
    const int row  = tile * 16 + m;
    const long long arow = (row < N) ? row : (N - 1);       // clamp, no exec dance

    v8f acc0 = {}, acc1 = {}, acc2 = {}, acc3 = {};
#pragma unroll
    for (int k = 0; k < FEAT_H; k += 4) {
        const float* ap = hagg + arow * FEAT_H + k + 2 * half;
        v2f a = { fmaxf(ap[0], 0.f), fmaxf(ap[1], 0.f) };
        const int kr0 = (k + 2 * half) * FEAT_OUT + m;
        const int kr1 = kr0 + FEAT_OUT;
        v2f b0 = { w2s[kr0 +  0], w2s[kr1 +  0] };
        v2f b1 = { w2s[kr0 + 16], w2s[kr1 + 16] };
        v2f b2 = { w2s[kr0 + 32], w2s[kr1 + 32] };
        v2f b3 = { w2s[kr0 + 48], w2s[kr1 + 48] };
        acc0 = __builtin_amdgcn_wmma_f32_16x16x4_f32(false, a, false, b0, (short)0, acc0, false, false);
        acc1 = __builtin_amdgcn_wmma_f32_16x16x4_f32(false, a, false, b1, (short)0, acc1, false, false);
        acc2 = __builtin_amdgcn_wmma_f32_16x16x4_f32(false, a, false, b2, (short)0, acc2, false, false);
        acc3 = __builtin_amdgcn_wmma_f32_16x16x4_f32(false, a, false, b3, (short)0, acc3, false, false);
    }
#pragma unroll
    for (int r = 0; r < 8; ++r) {
        int orow = tile * 16 + r + 8 * half;
        if (orow < N) {
            float* zp = z0 + (long long)orow * FEAT_OUT;
            zp[m +  0] = acc0[r];
            zp[m + 16] = acc1[r];
            zp[m + 32] = acc2[r];
            zp[m + 48] = acc3[r];
        }
    }
}

// ---------------------------------------------------------------------------
// out[dst] += in[src] * dinv[src]*dinv[dst]; items [0,E)=edges, [E,E+N)=loops.
template <int FEAT, int LOG2F>
__global__ void scatter_kernel(const float* __restrict__ in, float* __restrict__ out,
                               const int* __restrict__ ei, const float* __restrict__ dinv,
                               int E, int N) {
    long long t = (long long)blockIdx.x * blockDim.x + threadIdx.x;
    long long item = t >> LOG2F;
    int f = (int)(t & (FEAT - 1));
    if (item >= (long long)E + N) return;
    int s, d;
    if (item < E) { s = ei[item]; d = ei[E + item]; }
    else          { s = d = (int)(item - E); }
    float nrm = dinv[s] * dinv[d];
    unsafeAtomicAdd(&out[(long long)d * FEAT + f], in[(long long)s * FEAT + f] * nrm);
}

// ---------------------------------------------------------------------------
// Edge logits: 16 lanes/edge, butterfly reduce inside the 16-lane group.
__global__ void logits_kernel(const float* __restrict__ z,
                              const int* __restrict__ pos, const int* __restrict__ neg,
                              const float* __restrict__ We, const float* __restrict__ be,
                              float* __restrict__ out, int EP, int EN) {
    long long t = (long long)blockIdx.x * blockDim.x + threadIdx.x;
    long long g = t >> 4;
    int j = (int)(t & 15);
    if (g >= (long long)EP + EN) return;
    int s, d;
    if (g < EP) { s = pos[g];           d = pos[EP + g];           }
    else        { long long e = g - EP; s = neg[e]; d = neg[EN + e]; }
    const float* zs = z + (long long)s * FEAT_OUT;
    const float* zd = z + (long long)d * FEAT_OUT;
    float p = 0.f;
#pragma unroll
    for (int tt = 0; tt < 4; ++tt) {
        int c = j + 16 * tt;
        p += zs[c] * We[c] + zd[c] * We[FEAT_OUT + c];
    }
    p += __shfl_xor(p, 8);
    p += __shfl_xor(p, 4);
    p += __shfl_xor(p, 2);
    p += __shfl_xor(p, 1);
    if (j == 0) out[g] = p + be[0];
}

// ---------------------------------------------------------------------------
extern "C" void kernel_launch(void* const* d_in, const int* in_sizes, int n_in,
                              void* d_out, int out_size, void* d_ws, size_t ws_size,
                              hipStream_t stream) {
    const float* x   = (const float*)d_in[0];
    const float* W1  = (const float*)d_in[1];
    const float* b1  = (const float*)d_in[2];
    const float* W2  = (const float*)d_in[3];
    const float* b2  = (const float*)d_in[4];
    const float* We  = (const float*)d_in[5];
    const float* be  = (const float*)d_in[6];
    const int*   ei  = (const int*)d_in[7];
    const int*   pos = (const int*)d_in[8];
    const int*   neg = (const int*)d_in[9];

    const int N  = in_sizes[0] / FEAT_IN;   // 100000
    const int E  = in_sizes[7] / 2;         // 3200000
    const int EP = in_sizes[8] / 2;         // 1000000
    const int EN = in_sizes[9] / 2;         // 1000000

    float* ws    = (float*)d_ws;
    float* deg   = ws;                          // N
    float* dinv  = ws + (long long)N;           // N
    float* h0    = ws + 2LL  * N;               // N*16
    float* hagg  = ws + 18LL * N;               // N*16
    float* z0    = ws + 34LL * N;               // N*64
    float* z     = (float*)d_out;               // N*64
    float* logit = z + (long long)N * FEAT_OUT; // EP+EN

    const int B = 256;
    auto cdiv = [](long long a, long long b) { return (unsigned)((a + b - 1) / b); };

    init_deg_kernel<<<cdiv(N, B), B, 0, stream>>>(deg, N);
    deg_edge_kernel<<<cdiv(E, B), B, 0, stream>>>(ei, deg, E);
    dinv_kernel<<<cdiv(N, B), B, 0, stream>>>(deg, dinv, N);

    const int tiles = (N + 15) / 16;
    gemm1_kernel<<<cdiv(tiles, 8), B, 0, stream>>>(x, W1, h0, N);
    init_rep_kernel<<<cdiv((long long)N * FEAT_H, B), B, 0, stream>>>(hagg, b1, FEAT_H, (long long)N * FEAT_H);
    scatter_kernel<FEAT_H, 4><<<cdiv(((long long)E + N) * FEAT_H, B), B, 0, stream>>>(h0, hagg, ei, dinv, E, N);

    gemm2_kernel<<<cdiv(tiles, 8), B, 0, stream>>>(hagg, W2, z0, N);
    init_rep_kernel<<<cdiv((long long)N * FEAT_OUT, B), B, 0, stream>>>(z, b2, FEAT_OUT, (long long)N * FEAT_OUT);
    scatter_kernel<FEAT_OUT, 6><<<cdiv(((long long)E + N) * FEAT_OUT, B), B, 0, stream>>>(z0, z, ei, dinv, E, N);

    logits_kernel<<<cdiv(((long long)EP + EN) * 16, B), B, 0, stream>>>(z, pos, neg, We, be, logit, EP, EN);
}